// DataAugmentation_73289321939119
// MI455X (gfx1250) — compile-verified
//
#include <hip/hip_runtime.h>
#include <stdint.h>

#define SZ 64
#define PITCH 68                 // 68 % 64 banks == 4 -> conflict-free strided tile reads
#define CHS (SZ * PITCH)         // 4352 floats per channel plane in LDS
#define NPIX (SZ * SZ)

typedef __attribute__((ext_vector_type(2))) float v2f;
typedef __attribute__((ext_vector_type(8))) float v8f;
typedef __attribute__((__vector_size__(4 * sizeof(int)))) int v4i;

#define AS1 __attribute__((address_space(1)))
#define AS3 __attribute__((address_space(3)))

// ===================== CDNA5 async global<->LDS (ASYNCcnt path) =====================
__device__ __forceinline__ void async_ld_b128(const float* g, const float* l) {
#if __has_builtin(__builtin_amdgcn_global_load_async_to_lds_b128)
  __builtin_amdgcn_global_load_async_to_lds_b128((AS1 v4i*)g, (AS3 v4i*)l, 0, 0);
#else
  asm volatile("global_load_async_to_lds_b128 %0, %1, off"
               :: "v"((uint32_t)(uintptr_t)l), "v"(g) : "memory");
#endif
}
__device__ __forceinline__ void async_st_b128(float* g, const float* l) {
#if __has_builtin(__builtin_amdgcn_global_store_async_from_lds_b128)
  __builtin_amdgcn_global_store_async_from_lds_b128((AS1 v4i*)g, (AS3 v4i*)l, 0, 0);
#else
  asm volatile("global_store_async_from_lds_b128 %0, %1, off"
               :: "v"(g), "v"((uint32_t)(uintptr_t)l) : "memory");
#endif
}
__device__ __forceinline__ void wait_async0() {
#if __has_builtin(__builtin_amdgcn_s_wait_asynccnt)
  __builtin_amdgcn_s_wait_asynccnt(0);
#else
  asm volatile("s_wait_asynccnt 0" ::: "memory");
#endif
}

// ===================== exact JAX threefry2x32 PRNG =====================
struct KeyPair { uint32_t a, b; };

__device__ __forceinline__ void tf_round(uint32_t& x0, uint32_t& x1, int r) {
  x0 += x1;
  x1 = (x1 << r) | (x1 >> (32 - r));
  x1 ^= x0;
}
__device__ KeyPair tf2x32(KeyPair k, uint32_t c0, uint32_t c1) {
  uint32_t k0 = k.a, k1 = k.b, k2 = k.a ^ k.b ^ 0x1BD11BDAu;
  uint32_t x0 = c0 + k0, x1 = c1 + k1;
  tf_round(x0,x1,13); tf_round(x0,x1,15); tf_round(x0,x1,26); tf_round(x0,x1,6);
  x0 += k1; x1 += k2 + 1u;
  tf_round(x0,x1,17); tf_round(x0,x1,29); tf_round(x0,x1,16); tf_round(x0,x1,24);
  x0 += k2; x1 += k0 + 2u;
  tf_round(x0,x1,13); tf_round(x0,x1,15); tf_round(x0,x1,26); tf_round(x0,x1,6);
  x0 += k0; x1 += k1 + 3u;
  tf_round(x0,x1,17); tf_round(x0,x1,29); tf_round(x0,x1,16); tf_round(x0,x1,24);
  x0 += k1; x1 += k2 + 4u;
  tf_round(x0,x1,13); tf_round(x0,x1,15); tf_round(x0,x1,26); tf_round(x0,x1,6);
  x0 += k2; x1 += k0 + 5u;
  return {x0, x1};
}
// split(key, n): counts = iota(2n); x0 = first half, x1 = second half; out = concat(enc0, enc1)
__device__ void tf_split4(KeyPair k, KeyPair o[4]) {
  KeyPair p0 = tf2x32(k,0,4), p1 = tf2x32(k,1,5), p2 = tf2x32(k,2,6), p3 = tf2x32(k,3,7);
  o[0] = {p0.a, p1.a}; o[1] = {p2.a, p3.a}; o[2] = {p0.b, p1.b}; o[3] = {p2.b, p3.b};
}
__device__ void tf_split2(KeyPair k, KeyPair o[2]) {
  KeyPair p0 = tf2x32(k,0,2), p1 = tf2x32(k,1,3);
  o[0] = {p0.a, p1.a}; o[1] = {p0.b, p1.b};
}
__device__ void tf_split3(KeyPair k, KeyPair o[3]) {
  KeyPair p0 = tf2x32(k,0,3), p1 = tf2x32(k,1,4), p2 = tf2x32(k,2,5);
  o[0] = {p0.a, p1.a}; o[1] = {p2.a, p0.b}; o[2] = {p1.b, p2.b};
}
__device__ __forceinline__ KeyPair tf_fold_in(KeyPair k, uint32_t d) { return tf2x32(k, 0u, d); }
__device__ uint32_t tf_bits(KeyPair k, int b, int B) {   // random_bits of shape (B,) at index b
  int half = B >> 1;
  if (b < half) return tf2x32(k, (uint32_t)b, (uint32_t)(b + half)).a;
  return tf2x32(k, (uint32_t)(b - half), (uint32_t)b).b;
}
__device__ float tf_unif01(KeyPair k, int b, int B) {
  uint32_t bits = tf_bits(k, b, B);
  return __uint_as_float((bits >> 9) | 0x3f800000u) - 1.0f;
}
__device__ float tf_unif(KeyPair k, int b, int B, float lo, float hi) {
  float u = tf_unif01(k, b, B);
  return fmaxf(lo, u * (hi - lo) + lo);
}
__device__ int tf_randint10(KeyPair k, int b, int B) {   // JAX randint(0,10): two draws + remainder trick
  KeyPair ks[2]; tf_split2(k, ks);
  uint32_t hi = tf_bits(ks[0], b, B), lo = tf_bits(ks[1], b, B);
  return (int)(((hi % 10u) * 6u + (lo % 10u)) % 10u);   // multiplier = (2^32 mod 10) = 6
}

// ===================== per-image augmentation parameters =====================
__global__ void aug_params_kernel(float* __restrict__ P, int B) {
  int b = blockIdx.x * blockDim.x + threadIdx.x;
  if (b >= B) return;
  KeyPair root = {0u, 42u};
  KeyPair r4[4]; tf_split4(root, r4);
  KeyPair kflip = r4[0], kcrop = r4[1], kcj = r4[2], kra = r4[3];
  float* o = P + b * 16;
  o[0] = (tf_unif01(kflip, b, B) > 0.5f) ? 1.0f : 0.0f;
  // crop params: 10 rejection iterations
  float fh = 64.0f, fw = 64.0f, fi = 0.0f, fj = 0.0f;
  bool success = false;
  const float LO = -0.22314355131420971f;   // log(0.8)
  const float HI =  0.22314355131420976f;   // log(1.25)
  for (int it = 0; it < 10; ++it) {
    KeyPair kk = tf_fold_in(kcrop, (uint32_t)it);
    KeyPair s4[4]; tf_split4(kk, s4);
    float ta = 4096.0f * tf_unif(s4[0], b, B, 0.6f, 1.0f);
    float ar = expf(tf_unif(s4[1], b, B, LO, HI));
    float cw = rintf(sqrtf(ta * ar));
    float ch = rintf(sqrtf(ta / ar));
    bool valid = (cw > 0.0f) && (cw <= 64.0f) && (ch > 0.0f) && (ch <= 64.0f) && !success;
    float mi = fmaxf(64.0f - ch + 1.0f, 1.0f);
    float mj = fmaxf(64.0f - cw + 1.0f, 1.0f);
    float ri = floorf(tf_unif01(s4[2], b, B) * mi);
    float rj = floorf(tf_unif01(s4[3], b, B) * mj);
    if (valid) { fh = ch; fw = cw; fi = ri; fj = rj; success = true; }
  }
  o[1] = fi; o[2] = fj; o[3] = fh; o[4] = fw;
  KeyPair c4[4]; tf_split4(kcj, c4);
  o[5] = 1.0f + tf_unif(c4[0], b, B, -0.4f, 0.4f);
  o[6] = 1.0f + tf_unif(c4[1], b, B, -0.4f, 0.4f);
  o[7] = 1.0f + tf_unif(c4[2], b, B, -0.4f, 0.4f);
  o[8] = tf_unif(c4[3], b, B, -0.1f, 0.1f);
  for (int step = 0; step < 2; ++step) {
    KeyPair kk = tf_fold_in(kra, (uint32_t)step);
    KeyPair s3[3]; tf_split3(kk, s3);
    o[9 + step * 3]  = (float)tf_randint10(s3[0], b, B);
    o[10 + step * 3] = tf_unif01(s3[1], b, B);
    o[11 + step * 3] = tf_unif01(s3[2], b, B);
  }
  o[15] = 0.0f;
}

// ===================== helpers =====================
__device__ __forceinline__ float clip01(float v) { return fminf(fmaxf(v, 0.0f), 1.0f); }
__device__ __forceinline__ float grayf(float r, float g, float b) {
  return 0.2989f * r + 0.587f * g + 0.114f * b;
}
__device__ __forceinline__ int clampi(int v, int lo, int hi) {
  return v < lo ? lo : (v > hi ? hi : v);
}
__device__ float block_mean_gray(const float* img, float* red, int tid) {
  float s = 0.0f;
  for (int p = tid; p < NPIX; p += 256) {
    int o = (p >> 6) * PITCH + (p & 63);
    s += grayf(img[o], img[CHS + o], img[2 * CHS + o]);
  }
  red[tid] = s;
  __syncthreads();
  for (int off = 128; off > 0; off >>= 1) {
    if (tid < off) red[tid] += red[tid + off];
    __syncthreads();
  }
  float m = red[0] * (1.0f / 4096.0f);
  __syncthreads();
  return m;
}

// 64x64x64 fp32 GEMM on LDS operands via V_WMMA_F32_16X16X4_F32.
// D = A * B;  B_FROM_TRANSPOSE: B[k][n] = Bmat[n][k] (Bmat row-major, pitch 68).
// 16 output tiles, 2 per wave (8 wave32 waves). ISA 7.12.2 layouts:
//   A 16x4 : lane m = L%16, vgpr t holds K = 2*(L/16)+t
//   B 4x16 : lane n = L%16, vgpr t holds K = 2*(L/16)+t
//   D 16x16: lane n = L%16, vgpr v holds M = v + 8*(L/16)
template <bool B_FROM_TRANSPOSE>
__device__ void wmma_mm(const float* __restrict__ Amat, const float* __restrict__ Bmat,
                        float* __restrict__ Dmat, int tid) {
  const int lane = tid & 31;
  const int w = tid >> 5;
  const int m = lane & 15;
  const int h = lane >> 4;
  #pragma unroll
  for (int tt = 0; tt < 2; ++tt) {
    const int t = 2 * w + tt;
    const int j0 = (t >> 2) << 4;
    const int x0 = (t & 3) << 4;
    v8f acc = {};
    #pragma unroll
    for (int k0 = 0; k0 < 64; k0 += 4) {
      v2f a, bb;
      const float* ap = Amat + (j0 + m) * PITCH + (k0 + 2 * h);
      a.x = ap[0]; a.y = ap[1];
      if (B_FROM_TRANSPOSE) {
        const float* bp = Bmat + (x0 + m) * PITCH + (k0 + 2 * h);
        bb.x = bp[0]; bb.y = bp[1];
      } else {
        bb.x = Bmat[(k0 + 2 * h) * PITCH + (x0 + m)];
        bb.y = Bmat[(k0 + 2 * h + 1) * PITCH + (x0 + m)];
      }
      acc = __builtin_amdgcn_wmma_f32_16x16x4_f32(false, a, false, bb, (short)0, acc,
                                                  false, false);
    }
    float* dp = Dmat + (j0 + 8 * h) * PITCH + (x0 + m);
    #pragma unroll
    for (int v = 0; v < 8; ++v) dp[v * PITCH] = acc[v];
  }
}

// ===================== RandAugment branch (op uniform per block) =====================
__device__ void apply_aug(const float* __restrict__ src, float* __restrict__ dst,
                          int op, float u1, float u2, int tid, float* red) {
  if (op <= 4) {  // affine nearest: rotate / translate x / translate y / shear x / shear y
    float a00 = 1.0f, a01 = 0.0f, a10 = 0.0f, a11 = 1.0f, tx = 0.0f, ty = 0.0f;
    if (op == 0) {
      float ang = ((u1 > 0.5f) ? -1.0f : 1.0f) * 0.15707963267948966f;  // M*30deg
      float ca = cosf(ang), sa = sinf(ang);
      a00 = ca; a01 = sa; a10 = -sa; a11 = ca;
    } else if (op == 1) {
      tx = rintf(u2 * 18.0f - 9.0f);
    } else if (op == 2) {
      ty = rintf(u2 * 18.0f - 9.0f);
    } else if (op == 3) {
      a01 = 0.15838444032453627f;  // tan(9deg)
    } else {
      a10 = 0.15838444032453627f;
    }
    for (int p = tid; p < NPIX; p += 256) {
      float xs = (float)(p & 63), ys = (float)(p >> 6);
      float xo = xs - 31.5f - tx;
      float yo = ys - 31.5f - ty;
      float xi = a00 * xo + a01 * yo + 31.5f;
      float yi = a10 * xo + a11 * yo + 31.5f;
      int xn = (int)rintf(xi), yn = (int)rintf(yi);
      bool valid = (xn >= 0) && (xn < 64) && (yn >= 0) && (yn < 64);
      int so = clampi(yn, 0, 63) * PITCH + clampi(xn, 0, 63);
      int dOfs = (p >> 6) * PITCH + (p & 63);
      #pragma unroll
      for (int c = 0; c < 3; ++c)
        dst[c * CHS + dOfs] = valid ? src[c * CHS + so] : 0.0f;
    }
  } else if (op <= 7) {  // brightness / contrast / saturation
    float f = (float)(1.0 + (9.0 / 30.0) * 0.9);
    if (u1 > 0.5f) f = (float)(2.0 - (1.0 + (9.0 / 30.0) * 0.9));
    float mean = 0.0f;
    if (op == 6) mean = block_mean_gray(src, red, tid);
    for (int p = tid; p < NPIX; p += 256) {
      int o = (p >> 6) * PITCH + (p & 63);
      float r = src[o], g = src[CHS + o], b2 = src[2 * CHS + o];
      float rr, gg, bb;
      if (op == 5) {
        rr = clip01(r * f); gg = clip01(g * f); bb = clip01(b2 * f);
      } else if (op == 6) {
        rr = clip01(f * r + (1.0f - f) * mean);
        gg = clip01(f * g + (1.0f - f) * mean);
        bb = clip01(f * b2 + (1.0f - f) * mean);
      } else {
        float gr = grayf(r, g, b2);
        rr = clip01(f * r + (1.0f - f) * gr);
        gg = clip01(f * g + (1.0f - f) * gr);
        bb = clip01(f * b2 + (1.0f - f) * gr);
      }
      dst[o] = rr; dst[CHS + o] = gg; dst[2 * CHS + o] = bb;
    }
  } else if (op == 8) {  // solarize
    const float th = (float)(9.0 / 30.0);
    for (int p = tid; p < NPIX; p += 256) {
      int o = (p >> 6) * PITCH + (p & 63);
      #pragma unroll
      for (int c = 0; c < 3; ++c) {
        float v = src[c * CHS + o];
        dst[c * CHS + o] = (v > th) ? (1.0f - v) : v;
      }
    }
  } else {  // posterize, 32 levels
    for (int p = tid; p < NPIX; p += 256) {
      int o = (p >> 6) * PITCH + (p & 63);
      #pragma unroll
      for (int c = 0; c < 3; ++c) {
        float v = src[c * CHS + o];
        dst[c * CHS + o] = floorf(v * 31.0f) / 31.0f;
      }
    }
  }
}

// ===================== fused per-image pipeline: 1 workgroup = 1 image =====================
__global__ void __launch_bounds__(256)
aug_main_kernel(const float* __restrict__ in, const float* __restrict__ prm_all,
                float* __restrict__ out) {
  __shared__ float sA[3 * CHS];     // image buffer (input -> resampled -> jittered -> final)
  __shared__ float sScr[3 * CHS];   // [T | Wx | Wy] during resample; ping-pong buffer for aug
  __shared__ float sRed[256];

  const int b = blockIdx.x;
  const int tid = threadIdx.x;
  const float* prm = prm_all + (size_t)b * 16;

  // ---- issue async image load global -> LDS (row pitch 68 floats; all 16B aligned) ----
  {
    const float* gbase = in + (size_t)b * (3 * NPIX);
    #pragma unroll
    for (int k = 0; k < 12; ++k) {
      int ch = tid + (k << 8);                  // 3072 B128 chunks
      int c = ch >> 10, rem = ch & 1023;
      int row = rem >> 4, col = (rem & 15) << 2;
      async_ld_b128(gbase + c * NPIX + row * SZ + col,
                    sA + (c * SZ + row) * PITCH + col);
    }
  }

  // ---- build interpolation operators Wx/Wy while the loads are in flight ----
  float* sT  = sScr;
  float* sWx = sScr + CHS;
  float* sWy = sScr + 2 * CHS;
  for (int i = tid; i < 2 * CHS; i += 256) sWx[i] = 0.0f;
  __syncthreads();
  {
    const bool flip = prm[0] > 0.5f;
    const float fi = prm[1], fj = prm[2], fh = prm[3], fw = prm[4];
    if (tid < SZ) {               // Wx row = output column d (horizontal flip folded in)
      int d = tid;
      float sx = fminf(fmaxf((d + 0.5f) * fw / 64.0f - 0.5f, 0.0f), fw - 1.0f) + fj;
      float x0 = floorf(sx);
      float wx = sx - x0;
      int x0i = clampi((int)x0, 0, 63);
      int x1i = clampi(x0i + 1, 0, 63);
      if (flip) { x0i = 63 - x0i; x1i = 63 - x1i; }
      sWx[d * PITCH + x0i] = 1.0f - wx;
      sWx[d * PITCH + x1i] += wx;
    } else if (tid < 2 * SZ) {    // Wy row = output row d
      int d = tid - SZ;
      float sy = fminf(fmaxf((d + 0.5f) * fh / 64.0f - 0.5f, 0.0f), fh - 1.0f) + fi;
      float y0 = floorf(sy);
      float wy = sy - y0;
      int y0i = clampi((int)y0, 0, 63);
      int y1i = clampi(y0i + 1, 0, 63);
      sWy[d * PITCH + y0i] = 1.0f - wy;
      sWy[d * PITCH + y1i] += wy;
    }
  }
  wait_async0();
  __syncthreads();

  // ---- separable bilinear resized-crop as two fp32 WMMA GEMMs per channel ----
  for (int c = 0; c < 3; ++c) {
    wmma_mm<true >(sA + c * CHS, sWx, sT, tid);   // T    = img_c * Wx^T
    __syncthreads();
    wmma_mm<false>(sWy, sT, sA + c * CHS, tid);   // img_c = Wy * T
    __syncthreads();
  }

  // ---- color jitter in place on sA ----
  {
    const float bf = prm[5], cf = prm[6], sf = prm[7], hf = prm[8];
    float part = 0.0f;
    for (int p = tid; p < NPIX; p += 256) {       // brightness + partial gray sum
      int o = (p >> 6) * PITCH + (p & 63);
      float r = clip01(sA[o] * bf);
      float g = clip01(sA[CHS + o] * bf);
      float b2 = clip01(sA[2 * CHS + o] * bf);
      sA[o] = r; sA[CHS + o] = g; sA[2 * CHS + o] = b2;
      part += grayf(r, g, b2);
    }
    sRed[tid] = part;
    __syncthreads();
    for (int off = 128; off > 0; off >>= 1) {     // deterministic tree reduction
      if (tid < off) sRed[tid] += sRed[tid + off];
      __syncthreads();
    }
    const float mean = sRed[0] * (1.0f / 4096.0f);
    __syncthreads();
    for (int p = tid; p < NPIX; p += 256) {       // contrast -> saturation -> hue
      int o = (p >> 6) * PITCH + (p & 63);
      float r = sA[o], g = sA[CHS + o], b2 = sA[2 * CHS + o];
      r = clip01(cf * r + (1.0f - cf) * mean);
      g = clip01(cf * g + (1.0f - cf) * mean);
      b2 = clip01(cf * b2 + (1.0f - cf) * mean);
      float gr = grayf(r, g, b2);
      r = clip01(sf * r + (1.0f - sf) * gr);
      g = clip01(sf * g + (1.0f - sf) * gr);
      b2 = clip01(sf * b2 + (1.0f - sf) * gr);
      // rgb -> hsv (mirrors reference where-chains exactly)
      float maxc = fmaxf(r, fmaxf(g, b2));
      float minc = fminf(r, fminf(g, b2));
      bool eqc = (maxc == minc);
      float cr = maxc - minc;
      float sat = cr / (eqc ? 1.0f : maxc);
      float crd = eqc ? 1.0f : cr;
      float rc = (maxc - r) / crd;
      float gc = (maxc - g) / crd;
      float bc = (maxc - b2) / crd;
      float hr = (maxc == r) ? (bc - gc) : 0.0f;
      float hg = ((maxc == g) && (maxc != r)) ? (2.0f + rc - bc) : 0.0f;
      float hb = ((maxc != g) && (maxc != r)) ? (4.0f + gc - rc) : 0.0f;
      float hh = (hr + hg + hb) / 6.0f + 1.0f;
      float h = hh - floorf(hh);
      h = h + hf;
      h = h - floorf(h);
      // hsv -> rgb
      float h6 = h * 6.0f;
      float i6 = floorf(h6);
      float fr = h6 - i6;
      int i = ((int)i6) % 6;
      float v = maxc;
      float pc = clip01(v * (1.0f - sat));
      float qc = clip01(v * (1.0f - sat * fr));
      float tc = clip01(v * (1.0f - sat * (1.0f - fr)));
      float rr, gg, bb;
      switch (i) {
        case 0:  rr = v;  gg = tc; bb = pc; break;
        case 1:  rr = qc; gg = v;  bb = pc; break;
        case 2:  rr = pc; gg = v;  bb = tc; break;
        case 3:  rr = pc; gg = qc; bb = v;  break;
        case 4:  rr = tc; gg = pc; bb = v;  break;
        default: rr = v;  gg = pc; bb = qc; break;
      }
      sA[o] = rr; sA[CHS + o] = gg; sA[2 * CHS + o] = bb;
    }
    __syncthreads();
  }

  // ---- two RandAugment steps, ping-pong sA <-> sScr ----
  apply_aug(sA, sScr, (int)prm[9], prm[10], prm[11], tid, sRed);
  __syncthreads();
  apply_aug(sScr, sA, (int)prm[12], prm[13], prm[14], tid, sRed);
  __syncthreads();

  // ---- async store LDS -> global ----
  {
    float* gbase = out + (size_t)b * (3 * NPIX);
    #pragma unroll
    for (int k = 0; k < 12; ++k) {
      int ch = tid + (k << 8);
      int c = ch >> 10, rem = ch & 1023;
      int row = rem >> 4, col = (rem & 15) << 2;
      async_st_b128(gbase + c * NPIX + row * SZ + col,
                    sA + (c * SZ + row) * PITCH + col);
    }
    wait_async0();
  }
}

extern "C" void kernel_launch(void* const* d_in, const int* in_sizes, int n_in,
                              void* d_out, int out_size, void* d_ws, size_t ws_size,
                              hipStream_t stream) {
  (void)n_in; (void)out_size; (void)ws_size;
  const float* x = (const float*)d_in[0];
  float* out = (float*)d_out;
  const int B = in_sizes[0] / (3 * SZ * SZ);     // 1024
  float* params = (float*)d_ws;                  // 16 floats per image (64 KB)
  aug_params_kernel<<<(B + 255) / 256, 256, 0, stream>>>(params, B);
  aug_main_kernel<<<B, 256, 0, stream>>>(x, params, out);
}